// fm_13297218748808
// MI455X (gfx1250) — compile-verified
//
#include <hip/hip_runtime.h>
#include <math.h>

typedef __attribute__((ext_vector_type(2))) float v2f;
typedef __attribute__((ext_vector_type(4))) float f4;
typedef __attribute__((ext_vector_type(8))) float v8f;

typedef __attribute__((ext_vector_type(4))) int i4;
typedef __attribute__((address_space(1))) i4 g_i4;   // global-memory int4
typedef __attribute__((address_space(3))) i4 l_i4;   // LDS int4

#define BATCH   16384
#define NFIELD  10      // user, item, 8 feats
#define VEC     16
#define TD      160     // TOTAL_DIM
#define KK      32
#define DICT    10000

#define WAVES           4
#define ROWS_PER_WAVE   16
#define ROWS_PER_BLOCK  (WAVES * ROWS_PER_WAVE)   // 64
#define XSTRIDE         164                       // padded x-tile row stride (floats); 656B, 16B-aligned

// ---- CDNA5 async global->LDS copy (no VGPR round trip, tracked by ASYNCcnt) ----
__device__ __forceinline__ void async_copy16(const void* gsrc, void* ldst) {
#if __has_builtin(__builtin_amdgcn_global_load_async_to_lds_b128)
    __builtin_amdgcn_global_load_async_to_lds_b128(
        (g_i4*)gsrc, (l_i4*)ldst, 0, 0);
#else
    unsigned loff = (unsigned)(__UINTPTR_TYPE__)(l_i4*)ldst;
    unsigned long long ga = (unsigned long long)gsrc;
    asm volatile("global_load_async_to_lds_b128 %0, %1, off"
                 :: "v"(loff), "v"(ga) : "memory");
#endif
}

__device__ __forceinline__ void wait_async0() {
#if __has_builtin(__builtin_amdgcn_s_wait_asynccnt)
    __builtin_amdgcn_s_wait_asynccnt(0);
#else
    asm volatile("s_wait_asynccnt 0" ::: "memory");
#endif
}

__global__ __launch_bounds__(128) void fm_wmma_kernel(
    const int*   __restrict__ inputs,      // [BATCH][10] int32
    const float* __restrict__ user_table,  // [1000000][16]
    const float* __restrict__ item_table,  // [100000][16]
    const float* __restrict__ feat_tables, // [8][10000][16]
    const float* __restrict__ w,           // [160]
    const float* __restrict__ bias,        // [1]
    const float* __restrict__ k_mat,       // [160][32]
    float*       __restrict__ out)         // [BATCH]
{
    __shared__ alignas(16) float k_lds[TD * KK];                         // 20 KB
    __shared__ alignas(16) float w_lds[TD];                              // 640 B
    __shared__ alignas(16) float x_lds[WAVES][ROWS_PER_WAVE * XSTRIDE];  // 41 KB
    __shared__ alignas(16) float cross_lds[WAVES][16 * 16];              // 4 KB

    const int tid  = threadIdx.x;
    const int wave = tid >> 5;
    const int lane = tid & 31;
    const int tileRow0 = blockIdx.x * ROWS_PER_BLOCK + wave * ROWS_PER_WAVE;

    // ---- Stage k_mat and w into LDS via async global->LDS b128 ----
    for (int i = tid; i < TD * KK / 4; i += 128)
        async_copy16(k_mat + i * 4, k_lds + i * 4);
    if (tid < TD / 4)
        async_copy16(w + tid * 4, w_lds + tid * 4);

    // ---- Gather this wave's 16x160 x-tile straight into LDS (20 async b128 per lane) ----
    {
        float* xw = &x_lds[wave][0];
        #pragma unroll 4
        for (int t = 0; t < 20; ++t) {
            int u     = t * 32 + lane;      // 0..639 : 16 rows * 40 float4-chunks
            int r     = u / 40;             // tile row
            int c     = u - r * 40;         // chunk within row
            int field = c >> 2;             // 0..9
            int q     = c & 3;              // float4 within the 16-float vector
            int id    = inputs[(tileRow0 + r) * NFIELD + field];
            const float* src;
            if (field == 0)      src = user_table + (size_t)id * VEC;
            else if (field == 1) src = item_table + (size_t)id * VEC;
            else                 src = feat_tables + ((size_t)(field - 2) * DICT + (size_t)id) * VEC;
            async_copy16(src + q * 4, xw + r * XSTRIDE + field * VEC + q * 4);
        }
    }
    wait_async0();       // this wave's async LDS writes complete
    __syncthreads();     // make them visible block-wide

    // ---- Linear term: lane l handles row l&15; half l>>4 sums 80 cols ----
    const int rrow = lane & 15;
    const int half = lane >> 4;
    float lin;
    {
        float p = 0.f;
        const float* xr = &x_lds[wave][rrow * XSTRIDE];
        #pragma unroll 8
        for (int j = 0; j < 80; ++j) {
            int col = half * 80 + j;
            p += xr[col] * w_lds[col];
        }
        lin = p + __shfl_xor(p, 16, 32) + bias[0];
    }

    // ---- FM interaction via V_WMMA_F32_16X16X4_F32 ----
    // A frag (16x4 f32): lanes 0-15 hold M=lane,K={k0,k0+1}; lanes 16-31 hold K={k0+2,k0+3}
    // B frag (4x16 f32): lanes 0-15 hold K=k0,N=lane (v0) / K=k0+1 (v1); lanes 16-31 K=k0+2/k0+3
    float s[8];
    #pragma unroll
    for (int r = 0; r < 8; ++r) s[r] = 0.f;

    const float* xw   = &x_lds[wave][0];
    const int arow    = lane & 15;
    const int koff    = (lane >> 4) * 2;

    for (int nt = 0; nt < 2; ++nt) {
        v8f acc1 = {};
        v8f acc2 = {};
        const int ncol = nt * 16 + (lane & 15);
        #pragma unroll 4
        for (int k0 = 0; k0 < TD; k0 += 4) {
            v2f a = *(const v2f*)(xw + arow * XSTRIDE + k0 + koff);
            v2f bf;
            bf.x = k_lds[(k0 + koff) * KK + ncol];
            bf.y = k_lds[(k0 + koff + 1) * KK + ncol];
            v2f a2 = a * a;
            v2f b2 = bf * bf;
            acc1 = __builtin_amdgcn_wmma_f32_16x16x4_f32(false, a,  false, bf, (short)0, acc1, false, false);
            acc2 = __builtin_amdgcn_wmma_f32_16x16x4_f32(false, a2, false, b2, (short)0, acc2, false, false);
        }
        #pragma unroll
        for (int r = 0; r < 8; ++r) s[r] += acc1[r] * acc1[r] - acc2[r];
    }

    // ---- Per-row reduction of cross term through LDS scratch ----
    {
        float* cl = &cross_lds[wave][0];
        int hi = lane >> 4;
        int lo = lane & 15;
        #pragma unroll
        for (int r = 0; r < 8; ++r) cl[(r + 8 * hi) * 16 + lo] = s[r];
    }
    __syncthreads();

    if (lane < 16) {
        const float* cl = &cross_lds[wave][0];
        float c = 0.f;
        #pragma unroll
        for (int n = 0; n < 16; ++n) c += cl[lane * 16 + n];
        float z = lin + 0.5f * c;
        out[tileRow0 + lane] = 1.0f / (1.0f + __expf(-z));
    }
}

extern "C" void kernel_launch(void* const* d_in, const int* in_sizes, int n_in,
                              void* d_out, int out_size, void* d_ws, size_t ws_size,
                              hipStream_t stream) {
    const int*   inputs      = (const int*)d_in[0];
    const float* user_table  = (const float*)d_in[1];
    const float* item_table  = (const float*)d_in[2];
    const float* feat_tables = (const float*)d_in[3];
    const float* w           = (const float*)d_in[4];
    const float* bias        = (const float*)d_in[5];
    const float* k_mat       = (const float*)d_in[6];
    float*       out         = (float*)d_out;

    dim3 grid(BATCH / ROWS_PER_BLOCK);   // 256
    dim3 block(128);                     // 4 wave32
    fm_wmma_kernel<<<grid, block, 0, stream>>>(inputs, user_table, item_table,
                                               feat_tables, w, bias, k_mat, out);
}